// GATpy_49460843381580
// MI455X (gfx1250) — compile-verified
//
#include <hip/hip_runtime.h>
#include <math.h>

// Problem constants (match reference); N and E derived from in_sizes.
#define HIDF   128      // HID == ATOM
#define NHEAD  4
#define WHID   512      // NHEAD*HIDF (concat width)
#define NGROUP 64       // G
#define OUTD   256      // OUT
#define OUT2   128      // OUT/2
#define NCLSD  2
#define SLOPE  0.2f
#define KC     128      // K-chunk staged in LDS
#define BSTRIDE (KC + 2) // padded LDS stride (floats) to spread banks

typedef float v2f __attribute__((ext_vector_type(2)));
typedef float v8f __attribute__((ext_vector_type(8)));

__device__ __forceinline__ float lrelu(float x) { return x >= 0.f ? x : SLOPE * x; }

// Order-preserving float atomic max via int/uint atomics (works with -inf init).
__device__ __forceinline__ void atomicMaxF(float* addr, float val) {
  if (val >= 0.f) atomicMax((int*)addr, __float_as_int(val));
  else            atomicMin((unsigned int*)addr, __float_as_uint(val));
}

// ---------------------------------------------------------------------------
// fp32 WMMA GEMM, block-cooperative:
//   block = 8 waves = 128 output rows x 64 output cols
//   wave  = 16 rows x 64 cols (4 x v8f accumulators)
//   K processed in KC=128 chunks; B panel (KC x 64) staged in LDS transposed
//   (Bs[col][k], padded stride) so each B fragment is one aligned ds_load_b64.
// WMMA mapping (V_WMMA_F32_16X16X4_F32, ISA 7.12.2):
//   A 16x4: lane l / l+16 holds row l, K={0,1}/{2,3}   -> v2f (64-bit load)
//   B 4x16: lane l / l+16 holds col l, K={0,1}/{2,3}   -> v2f (ds_load_b64)
//   C/D:    vgpr i, lanes 0-15 -> row i, lanes 16-31 -> row i+8
// No early returns: ragged waves clamp loads / guard stores so barriers and
// the EXEC-all-ones WMMA requirement hold for every wave in the block.
// ---------------------------------------------------------------------------
__global__ void wmma_gemm(const float* __restrict__ A, const float* __restrict__ B,
                          const float* __restrict__ bias, float* __restrict__ C,
                          int M, int K, int Nc) {
  __shared__ float Bs[64 * BSTRIDE];           // 33.3 KB of the 320 KB WGP LDS

  const int nslab = Nc >> 6;                   // 64-col slabs
  const int mblk  = blockIdx.x / nslab;
  const int nblk  = blockIdx.x - mblk * nslab;
  const int wave  = threadIdx.x >> 5;          // 0..7
  const int lane  = threadIdx.x & 31;
  const int half  = lane >> 4;                 // 0: lanes 0-15, 1: lanes 16-31
  const int l     = lane & 15;

  const int mt    = mblk * 8 + wave;           // this wave's 16-row tile
  const int arow  = min(mt * 16 + l, M - 1);   // clamped (ragged tail)
  const int col0  = nblk * 64;

  // staging coords: 4 row-groups x 64 cols
  const int sc = threadIdx.x & 63;             // B column within slab
  const int sr = threadIdx.x >> 6;             // 0..3 -> k rows sr, sr+4, ...

  const float* Ap = A + (size_t)arow * K + 2 * half;

  v8f acc0 = {}, acc1 = {}, acc2 = {}, acc3 = {};

  for (int k0 = 0; k0 < K; k0 += KC) {
    const int kc = min(KC, K - k0);
    __syncthreads();                           // LDS WAR vs previous chunk
    for (int kk = sr; kk < kc; kk += 4)        // coalesced 64-wide row reads
      Bs[sc * BSTRIDE + kk] = B[(size_t)(k0 + kk) * Nc + col0 + sc];
    __syncthreads();

    __builtin_prefetch(Ap + k0 + KC, 0, 1);    // next A chunk -> global_prefetch_b8

    const float* Bl0 = &Bs[(l +  0) * BSTRIDE + 2 * half];
    const float* Bl1 = &Bs[(l + 16) * BSTRIDE + 2 * half];
    const float* Bl2 = &Bs[(l + 32) * BSTRIDE + 2 * half];
    const float* Bl3 = &Bs[(l + 48) * BSTRIDE + 2 * half];
    for (int k = 0; k < kc; k += 4) {
      v2f a  = *(const v2f*)(Ap + k0 + k);     // one A fragment feeds 4 WMMAs
      v2f b0 = *(const v2f*)(Bl0 + k);
      v2f b1 = *(const v2f*)(Bl1 + k);
      v2f b2 = *(const v2f*)(Bl2 + k);
      v2f b3 = *(const v2f*)(Bl3 + k);
      acc0 = __builtin_amdgcn_wmma_f32_16x16x4_f32(false, a, false, b0, (short)0, acc0, false, false);
      acc1 = __builtin_amdgcn_wmma_f32_16x16x4_f32(false, a, false, b1, (short)0, acc1, false, false);
      acc2 = __builtin_amdgcn_wmma_f32_16x16x4_f32(false, a, false, b2, (short)0, acc2, false, false);
      acc3 = __builtin_amdgcn_wmma_f32_16x16x4_f32(false, a, false, b3, (short)0, acc3, false, false);
    }
  }

  // epilogue: bias + guarded stores (rows >= M dropped)
  v8f* accs[4] = {&acc0, &acc1, &acc2, &acc3};
#pragma unroll
  for (int j = 0; j < 4; ++j) {
    const int ccol = col0 + 16 * j + l;
    const float bv = bias ? bias[ccol] : 0.f;
#pragma unroll
    for (int i = 0; i < 8; ++i) {
      const int row = mt * 16 + i + 8 * half;
      if (row < M) C[(size_t)row * Nc + ccol] = (*accs[j])[i] + bv;
    }
  }
}

// s_src[n,h] = <h[n,h,:], a_s[h,:]>, s_dst likewise. idx == node*NHEAD+head.
__global__ void attn_scores(const float* __restrict__ h, const float* __restrict__ a_s,
                            const float* __restrict__ a_d, float* __restrict__ ssrc,
                            float* __restrict__ sdst, int n) {
  int idx = blockIdx.x * blockDim.x + threadIdx.x;
  if (idx >= n * NHEAD) return;
  int hd = idx & (NHEAD - 1);
  const float* hp = h + (size_t)idx * HIDF;
  const float* as = a_s + hd * HIDF;
  const float* ad = a_d + hd * HIDF;
  float s1 = 0.f, s2 = 0.f;
  for (int c = 0; c < HIDF; c += 4) {
    float4 hv = *(const float4*)(hp + c);
    float4 av = *(const float4*)(as + c);
    float4 dv = *(const float4*)(ad + c);
    s1 += hv.x * av.x + hv.y * av.y + hv.z * av.z + hv.w * av.w;
    s2 += hv.x * dv.x + hv.y * dv.y + hv.z * dv.z + hv.w * dv.w;
  }
  ssrc[idx] = s1;
  sdst[idx] = s2;
}

__global__ void fill_f32(float* __restrict__ p, float v, int n) {
  int i = blockIdx.x * blockDim.x + threadIdx.x;
  if (i < n) p[i] = v;
}

// e[e,h] = lrelu(ssrc[src]+sdst[dst]); segment max into m[dst,h].
__global__ void edge_logits_max(const int* __restrict__ ei, int E, int n,
                                const float* __restrict__ ssrc,
                                const float* __restrict__ sdst,
                                float* __restrict__ eatt, float* __restrict__ m) {
  int idx = blockIdx.x * blockDim.x + threadIdx.x;
  if (idx >= (E + n) * NHEAD) return;
  int e = idx >> 2, hd = idx & 3;
  int s, d;
  if (e < E) { s = ei[e]; d = ei[E + e]; } else { s = e - E; d = s; }
  float v = lrelu(ssrc[s * NHEAD + hd] + sdst[d * NHEAD + hd]);
  eatt[idx] = v;
  atomicMaxF(&m[d * NHEAD + hd], v);
}

__global__ void fix_m(float* __restrict__ m, int count) {
  int i = blockIdx.x * blockDim.x + threadIdx.x;
  if (i >= count) return;
  float v = m[i];
  if (!(v > -3.0e38f && v < 3.0e38f)) m[i] = 0.f;   // mimic isfinite() guard
}

// ex = exp(e - m[dst]); den[dst,h] += ex (eatt overwritten in place with ex).
__global__ void edge_expsum(const int* __restrict__ ei, int E, int n,
                            float* __restrict__ eatt, const float* __restrict__ m,
                            float* __restrict__ den) {
  int idx = blockIdx.x * blockDim.x + threadIdx.x;
  if (idx >= (E + n) * NHEAD) return;
  int e = idx >> 2, hd = idx & 3;
  int d = (e < E) ? ei[E + e] : (e - E);
  float ex = expf(eatt[idx] - m[d * NHEAD + hd]);
  eatt[idx] = ex;
  atomicAdd(&den[d * NHEAD + hd], ex);
}

// One wave32 per (edge,head): alpha = ex/den[dst]; acc[dst,h,:] += alpha*h[src,h,:].
__global__ void edge_aggr(const int* __restrict__ ei, int E, int n,
                          const float* __restrict__ ex, const float* __restrict__ den,
                          const float* __restrict__ h, float* __restrict__ acc,
                          float* __restrict__ alpha_out) {
  long gid  = (long)blockIdx.x * blockDim.x + threadIdx.x;
  long wave = gid >> 5;
  int  lane = threadIdx.x & 31;
  if (wave >= (long)(E + n) * NHEAD) return;
  int e = (int)(wave >> 2), hd = (int)(wave & 3);
  int s, d;
  if (e < E) { s = ei[e]; d = ei[E + e]; } else { s = e - E; d = s; }
  float a = ex[wave] / den[(size_t)d * NHEAD + hd];
  if (alpha_out && lane == 0) alpha_out[wave] = a;
  const float* hp = h + ((size_t)s * NHEAD + hd) * HIDF + lane * 4;
  float*       ap = acc + ((size_t)d * NHEAD + hd) * HIDF + lane * 4;
  float4 hv = *(const float4*)hp;
  atomicAdd(ap + 0, a * hv.x);
  atomicAdd(ap + 1, a * hv.y);
  atomicAdd(ap + 2, a * hv.z);
  atomicAdd(ap + 3, a * hv.w);
}

// Layer1 epilogue: head mean + bias, LayerNorm, leaky ReLU. blockDim == HIDF.
__global__ void l1_post(const float* __restrict__ acc, const float* __restrict__ b1,
                        const float* __restrict__ g, const float* __restrict__ b,
                        float* __restrict__ x2) {
  __shared__ float sh[HIDF];
  int nd = blockIdx.x, c = threadIdx.x;
  const float* ap = acc + (size_t)nd * WHID;
  float v = 0.25f * (ap[c] + ap[HIDF + c] + ap[2 * HIDF + c] + ap[3 * HIDF + c]) + b1[c];
  sh[c] = v; __syncthreads();
  for (int s = HIDF / 2; s > 0; s >>= 1) { if (c < s) sh[c] += sh[c + s]; __syncthreads(); }
  float mu = sh[0] * (1.0f / HIDF);
  __syncthreads();
  float dv = v - mu;
  sh[c] = dv * dv; __syncthreads();
  for (int s = HIDF / 2; s > 0; s >>= 1) { if (c < s) sh[c] += sh[c + s]; __syncthreads(); }
  float var = sh[0] * (1.0f / HIDF);
  float y = g[c] * dv * rsqrtf(var + 1e-5f) + b[c];
  x2[(size_t)nd * HIDF + c] = lrelu(y);
}

// Layer2 epilogue: bias + leaky ReLU + mean-pool accumulate. blockDim == WHID.
__global__ void l2_post_pool(const float* __restrict__ acc, const float* __restrict__ b2,
                             const int* __restrict__ batch, float* __restrict__ pooled,
                             float* __restrict__ cnt) {
  int nd = blockIdx.x, c = threadIdx.x;
  float y = lrelu(acc[(size_t)nd * WHID + c] + b2[c]);
  int gidx = batch[nd];
  atomicAdd(&pooled[(size_t)gidx * WHID + c], y);
  if (c == 0) atomicAdd(&cnt[gidx], 1.0f);
}

__global__ void pool_div(float* __restrict__ pooled, const float* __restrict__ cnt) {
  int i = blockIdx.x * blockDim.x + threadIdx.x;
  if (i >= NGROUP * WHID) return;
  pooled[i] /= fmaxf(cnt[i / WHID], 1.0f);
}

// Generic per-row LayerNorm + ReLU. blockDim == dim (power of two), dynamic LDS.
__global__ void ln_relu(float* __restrict__ x, const float* __restrict__ g,
                        const float* __restrict__ b, int dim) {
  extern __shared__ float sh[];
  int r = blockIdx.x, c = threadIdx.x;
  float v = x[(size_t)r * dim + c];
  sh[c] = v; __syncthreads();
  for (int s = dim >> 1; s > 0; s >>= 1) { if (c < s) sh[c] += sh[c + s]; __syncthreads(); }
  float mu = sh[0] / dim;
  __syncthreads();
  float dv = v - mu;
  sh[c] = dv * dv; __syncthreads();
  for (int s = dim >> 1; s > 0; s >>= 1) { if (c < s) sh[c] += sh[c + s]; __syncthreads(); }
  float var = sh[0] / dim;
  float y = g[c] * dv * rsqrtf(var + 1e-5f) + b[c];
  x[(size_t)r * dim + c] = fmaxf(y, 0.f);
}

// Final 128->2 logits + softmax, one thread per graph.
__global__ void head_out(const float* __restrict__ t2, const float* __restrict__ Ws3,
                         const float* __restrict__ bs3, float* __restrict__ out) {
  int r = blockIdx.x * blockDim.x + threadIdx.x;
  if (r >= NGROUP) return;
  float l0 = bs3[0], l1 = bs3[1];
  for (int c = 0; c < OUT2; ++c) {
    float v = t2[r * OUT2 + c];
    l0 += v * Ws3[c * 2 + 0];
    l1 += v * Ws3[c * 2 + 1];
  }
  float mx = fmaxf(l0, l1);
  float e0 = expf(l0 - mx), e1 = expf(l1 - mx), s = e0 + e1;
  out[r * 2 + 0] = e0 / s;
  out[r * 2 + 1] = e1 / s;
}

extern "C" void kernel_launch(void* const* d_in, const int* in_sizes, int n_in,
                              void* d_out, int out_size, void* d_ws, size_t ws_size,
                              hipStream_t stream) {
  (void)n_in; (void)out_size; (void)ws_size;
  const float* x    = (const float*)d_in[0];
  const int*   ei   = (const int*)d_in[1];
  const int*   batch= (const int*)d_in[2];
  const float* W1   = (const float*)d_in[3];
  const float* as1  = (const float*)d_in[4];
  const float* ad1  = (const float*)d_in[5];
  const float* b1   = (const float*)d_in[6];
  const float* g_ln = (const float*)d_in[7];
  const float* b_ln = (const float*)d_in[8];
  const float* W2   = (const float*)d_in[9];
  const float* as2  = (const float*)d_in[10];
  const float* ad2  = (const float*)d_in[11];
  const float* b2   = (const float*)d_in[12];
  const float* Wl0  = (const float*)d_in[13];
  const float* bl0  = (const float*)d_in[14];
  const float* Ws1  = (const float*)d_in[15];
  const float* bs1  = (const float*)d_in[16];
  const float* g1   = (const float*)d_in[17];
  const float* bb1  = (const float*)d_in[18];
  const float* Ws2  = (const float*)d_in[19];
  const float* bs2  = (const float*)d_in[20];
  const float* g2   = (const float*)d_in[21];
  const float* bb2  = (const float*)d_in[22];
  const float* Ws3  = (const float*)d_in[23];
  const float* bs3  = (const float*)d_in[24];

  const int N  = in_sizes[0] / HIDF;   // ATOM == 128
  const int E  = in_sizes[1] / 2;
  const int TE = E + N;
  const int NH = N * NHEAD;
  const int TEH = TE * NHEAD;

  float* out       = (float*)d_out;
  float* alpha_out = out + NGROUP * NCLSD;   // [TE, NHEAD] row-major

  // Workspace carve-up (h and acc reused across both GAT layers).
  size_t off = 0;
  auto alloc = [&](size_t bytes) -> float* {
    off = (off + 255) & ~(size_t)255;
    float* p = (float*)((char*)d_ws + off);
    off += bytes;
    return p;
  };
  float* h      = alloc((size_t)N * WHID * 4);
  float* acc    = alloc((size_t)N * WHID * 4);
  float* x2     = alloc((size_t)N * HIDF * 4);
  float* ssrc   = alloc((size_t)NH * 4);
  float* sdst   = alloc((size_t)NH * 4);
  float* eatt   = alloc((size_t)TEH * 4);
  float* mbuf   = alloc((size_t)NH * 4);
  float* den    = alloc((size_t)NH * 4);
  float* pooled = alloc((size_t)NGROUP * WHID * 4);
  float* cnt    = alloc((size_t)NGROUP * 4);
  float* z0     = alloc((size_t)NGROUP * HIDF * 4);
  float* t1     = alloc((size_t)NGROUP * OUTD * 4);
  float* t2     = alloc((size_t)NGROUP * OUT2 * 4);

  auto cdiv = [](long a, long b) { return (int)((a + b - 1) / b); };
  auto gemm = [&](const float* A, const float* B, const float* bias, float* C,
                  int M, int K, int Nc) {
    int mtiles  = (M + 15) / 16;
    int mblocks = (mtiles + 7) / 8;      // 8 waves (m-tiles) per block
    int nslab   = Nc / 64;               // 64-col slab per block
    wmma_gemm<<<mblocks * nslab, 256, 0, stream>>>(A, B, bias, C, M, K, Nc);
  };

  // ===================== GAT layer 1 (concat=False) =====================
  gemm(x, W1, nullptr, h, N, HIDF, WHID);
  attn_scores<<<cdiv(NH, 256), 256, 0, stream>>>(h, as1, ad1, ssrc, sdst, N);
  fill_f32<<<cdiv(NH, 256), 256, 0, stream>>>(mbuf, -INFINITY, NH);
  fill_f32<<<cdiv(NH, 256), 256, 0, stream>>>(den, 0.f, NH);
  fill_f32<<<cdiv((long)N * WHID, 256), 256, 0, stream>>>(acc, 0.f, N * WHID);
  edge_logits_max<<<cdiv(TEH, 256), 256, 0, stream>>>(ei, E, N, ssrc, sdst, eatt, mbuf);
  fix_m<<<cdiv(NH, 256), 256, 0, stream>>>(mbuf, NH);
  edge_expsum<<<cdiv(TEH, 256), 256, 0, stream>>>(ei, E, N, eatt, mbuf, den);
  edge_aggr<<<cdiv((long)TEH * 32, 256), 256, 0, stream>>>(ei, E, N, eatt, den, h, acc,
                                                           nullptr);
  l1_post<<<N, HIDF, 0, stream>>>(acc, b1, g_ln, b_ln, x2);

  // ===================== GAT layer 2 (concat=True) ======================
  gemm(x2, W2, nullptr, h, N, HIDF, WHID);
  attn_scores<<<cdiv(NH, 256), 256, 0, stream>>>(h, as2, ad2, ssrc, sdst, N);
  fill_f32<<<cdiv(NH, 256), 256, 0, stream>>>(mbuf, -INFINITY, NH);
  fill_f32<<<cdiv(NH, 256), 256, 0, stream>>>(den, 0.f, NH);
  fill_f32<<<cdiv((long)N * WHID, 256), 256, 0, stream>>>(acc, 0.f, N * WHID);
  edge_logits_max<<<cdiv(TEH, 256), 256, 0, stream>>>(ei, E, N, ssrc, sdst, eatt, mbuf);
  fix_m<<<cdiv(NH, 256), 256, 0, stream>>>(mbuf, NH);
  edge_expsum<<<cdiv(TEH, 256), 256, 0, stream>>>(ei, E, N, eatt, mbuf, den);
  edge_aggr<<<cdiv((long)TEH * 32, 256), 256, 0, stream>>>(ei, E, N, eatt, den, h, acc,
                                                           alpha_out);

  // ========================= pooling + head ============================
  fill_f32<<<cdiv(NGROUP * WHID, 256), 256, 0, stream>>>(pooled, 0.f, NGROUP * WHID);
  fill_f32<<<1, NGROUP, 0, stream>>>(cnt, 0.f, NGROUP);
  l2_post_pool<<<N, WHID, 0, stream>>>(acc, b2, batch, pooled, cnt);
  pool_div<<<cdiv(NGROUP * WHID, 256), 256, 0, stream>>>(pooled, cnt);

  gemm(pooled, Wl0, bl0, z0, NGROUP, WHID, HIDF);              // [64,512]x[512,128]
  gemm(z0, Ws1, bs1, t1, NGROUP, HIDF, OUTD);                  // [64,128]x[128,256]
  ln_relu<<<NGROUP, OUTD, OUTD * sizeof(float), stream>>>(t1, g1, bb1, OUTD);
  gemm(t1, Ws2, bs2, t2, NGROUP, OUTD, OUT2);                  // [64,256]x[256,128]
  ln_relu<<<NGROUP, OUT2, OUT2 * sizeof(float), stream>>>(t2, g2, bb2, OUT2);
  head_out<<<1, NGROUP, 0, stream>>>(t2, Ws3, bs3, out);
}